// RationalRMSNorm_48060684042972
// MI455X (gfx1250) — compile-verified
//
#include <hip/hip_runtime.h>
#include <hip/hip_bf16.h>
#include <stdint.h>

// RMSNorm, fp32, rows of H=4096, B*S=16384 rows.
// Memory-bound: ~512 MiB traffic -> ~22.5 us roofline at 23.3 TB/s.
// Strategy: one workgroup per row; async-copy the 16 KB row into LDS
// (gfx1250 GLOBAL_LOAD_ASYNC_TO_LDS_B128 with NT hint, ASYNCcnt wait),
// single ds_load_b128 pass into registers, wave32 butterfly + LDS cross-wave
// reduction, rsqrt + 1 Newton step, then non-temporal b128 stores.
// x/out are streamed once -> NT keeps the reused weight vector in L2.

#define H_DIM 4096
#define TPB   256
#define WAVES (TPB / 32)                    // 8 waves per block (wave32)
#define V4PT  (H_DIM / (TPB * 4))           // 4 float4 chunks per thread

typedef __attribute__((address_space(3))) float lds_float;
typedef float v4f __attribute__((ext_vector_type(4)));

__global__ __launch_bounds__(TPB)
void rmsnorm_f32_kernel(const float* __restrict__ x,
                        const float* __restrict__ w,
                        float* __restrict__ out) {
    __shared__ float s_x[H_DIM];            // 16 KB row staging
    __shared__ float s_red[WAVES];

    const int    tid  = (int)threadIdx.x;
    const size_t row  = (size_t)blockIdx.x;
    const float* __restrict__ xrow = x   + row * (size_t)H_DIM;
    float*       __restrict__ orow = out + row * (size_t)H_DIM;

    // ---- Stage row into LDS via gfx1250 async-to-LDS copies (ASYNCcnt) ----
#if defined(__AMDGCN__)
    {
        // LDS byte address of s_x (addrspacecast generic -> LDS strips aperture)
        uint32_t lds_base = (uint32_t)(size_t)(lds_float*)s_x;
#pragma unroll
        for (int i = 0; i < V4PT; ++i) {
            uint32_t byte_off = (uint32_t)(tid + i * TPB) * 16u;  // b128 per lane
            uint32_t lds_addr = lds_base + byte_off;
            // GVS mode: mem = SGPR64(saddr) + VGPR32(vaddr); vdst = LDS address.
            // NT hint: row is streamed exactly once -> don't thrash L2.
            asm volatile("global_load_async_to_lds_b128 %0, %1, %2 th:TH_LOAD_NT"
                         :
                         : "v"(lds_addr), "v"(byte_off), "s"(xrow)
                         : "memory");
        }
        asm volatile("s_wait_asynccnt 0x0" ::: "memory");
    }
#else
    for (int i = 0; i < V4PT; ++i) {
        int col = (tid + i * TPB) * 4;
        *(float4*)&s_x[col] = *(const float4*)&xrow[col];
    }
#endif
    __syncthreads();

    // ---- Single LDS pass: pull 16 values to registers, sum of squares ----
    v4f   v[V4PT];
    float sum = 0.f;
#pragma unroll
    for (int i = 0; i < V4PT; ++i) {
        int col = (tid + i * TPB) * 4;
        v[i] = *(const v4f*)&s_x[col];         // ds_load_b128
        sum  = fmaf(v[i].x, v[i].x, sum);
        sum  = fmaf(v[i].y, v[i].y, sum);
        sum  = fmaf(v[i].z, v[i].z, sum);
        sum  = fmaf(v[i].w, v[i].w, sum);
    }

    // ---- wave32 butterfly reduction + cross-wave combine via LDS ----
#pragma unroll
    for (int m = 16; m >= 1; m >>= 1)
        sum += __shfl_xor(sum, m, 32);
    if ((tid & 31) == 0) s_red[tid >> 5] = sum;
    __syncthreads();
    float total = 0.f;
#pragma unroll
    for (int i = 0; i < WAVES; ++i) total += s_red[i];  // broadcast reads

    // ---- variance -> inv rms (clip 1e-8, +eps, rsqrt + one Newton step) ----
    float variance = total * (1.0f / (float)H_DIM);
    variance = fmaxf(variance, 1e-8f);
    float a = variance + 1e-6f;
    float y = rsqrtf(a);                        // v_rsq_f32
    y = y * fmaf(-0.5f * a, y * y, 1.5f);       // NR step -> full fp32 precision

    // ---- normalize, apply weight, stream out (non-temporal b128 stores) ----
#pragma unroll
    for (int i = 0; i < V4PT; ++i) {
        int col = (tid + i * TPB) * 4;
        const v4f wv = *(const v4f*)&w[col];    // reused across rows: keep RT
        v4f o;
        o.x = v[i].x * y * wv.x;
        o.y = v[i].y * y * wv.y;
        o.z = v[i].z * y * wv.z;
        o.w = v[i].w * y * wv.w;
        __builtin_nontemporal_store(o, (v4f*)&orow[col]);  // global_store_b128 NT
    }
}

extern "C" void kernel_launch(void* const* d_in, const int* in_sizes, int n_in,
                              void* d_out, int out_size, void* d_ws, size_t ws_size,
                              hipStream_t stream) {
    const float* x = (const float*)d_in[0];   // [B, S, H] fp32
    const float* w = (const float*)d_in[1];   // [H] fp32
    float* out = (float*)d_out;               // [B, S, H] fp32

    const int rows = in_sizes[0] / H_DIM;     // B*S = 16384
    rmsnorm_f32_kernel<<<rows, TPB, 0, stream>>>(x, w, out);
}